// PerBandSoftAGC_44461501448599
// MI455X (gfx1250) — compile-verified
//
#include <hip/hip_runtime.h>
#include <hip/hip_bf16.h>
#include <math.h>

// ---------------------------------------------------------------------------
// PerBandSoftAGC on MI455X (gfx1250, wave32)
//
// out[b,c,t] = x[b,c,t] / ( clip(M,1e-6)^alpha[c] + 0.1 )
// M[b,c,t]   = sum_{j=0..99} w[c][j] * |x[b,c,t-j]|      (causal, zero-padded)
// w[c][j]    = s(1-s)^j / (1-(1-s)^100 + 1e-8),  s = sigmoid(log_s[c])
// alpha[c]   = 0.5 * sigmoid(alpha_raw[c])
//
// FIR -> Toeplitz matmul on V_WMMA_F32_16X16X4_F32 (fp32, matches reference).
// LDS staging via Tensor Data Mover (tensor_load_to_lds) with pad_enable
// reproducing the bank-conflict padding XI(i)=i+(i>>4); TDM tensor_dim0 OOB
// rule zero-fills the series tail for free.
// ---------------------------------------------------------------------------

typedef float v2f __attribute__((ext_vector_type(2)));
typedef float v8f __attribute__((ext_vector_type(8)));
typedef unsigned int u32x4 __attribute__((ext_vector_type(4)));
typedef int i32x4 __attribute__((ext_vector_type(4)));
typedef int i32x8 __attribute__((ext_vector_type(8)));

#define T_LEN   8192
#define KC_CH   128          // 8 bands * 16 channels
#define NSERIES 4096         // 32 * 128
#define WIN     2048         // outputs per block
#define NWIN    4            // T_LEN / WIN
#define HALO    99           // FIR history
#define HALOP   112          // staged halo, multiple of 16 so TDM pad phase == XI
#define BOFF    (HALOP - HALO)   // 13: extra shift of B reads in LDS
#define NXRAW   2161         // raw indices used: 0 .. 1792+13+115+240 = 2160
#define XI(i)   ((i) + ((i) >> 4))   // LDS pad: stride-16 gathers hit 64 banks conflict-free
#define LDSX_SZ 2304         // >= XI(2160)+1 = 2296

#if defined(__has_builtin)
# if __has_builtin(__builtin_amdgcn_tensor_load_to_lds) && \
     __has_builtin(__builtin_amdgcn_s_wait_tensorcnt)
#  define USE_TDM 1
# endif
#endif
#ifndef USE_TDM
# define USE_TDM 0
#endif

__global__ __launch_bounds__(256)
void agc_wmma_kernel(const float* __restrict__ x,
                     const float* __restrict__ alpha_raw,
                     const float* __restrict__ log_s,
                     float* __restrict__ out)
{
    __shared__ float lds_x[LDSX_SZ];     // raw x window [wbase-112, ...], XI-padded layout
    __shared__ float lds_w[132];         // wrev with guard pads: [16+m] = w[99-m], else 0
    __shared__ float lds_o[WIN];         // output staging for coalesced stores

    const int tid    = threadIdx.x;
    const int series = blockIdx.x / NWIN;          // b*128 + c
    const int win    = blockIdx.x - series * NWIN;
    const int c      = series & (KC_CH - 1);
    const int wbase  = win * WIN;                  // first output t of this window

    // ---- per-channel scalars (uniform; SGPR loads) ----
    const float s     = 1.0f / (1.0f + expf(-log_s[c]));
    const float norm  = 1.0f - powf(1.0f - s, 100.0f) + 1e-8f;  // closed-form sum
    const float alpha = 0.5f / (1.0f + expf(-alpha_raw[c]));

    // ---- build reversed+padded weights in LDS (disjoint writes, one pass) ----
    if (tid < 100) {
        // wrev[m] = impulse[99-m] / norm,  stored at lds_w[16+m]
        lds_w[16 + tid] = s * powf(1.0f - s, (float)(99 - tid)) / norm;
    } else if (tid < 132) {
        int z = tid - 100;                          // 32 guard entries
        lds_w[(z < 16) ? z : (z + 100)] = 0.0f;     // [0..15] and [116..131]
    }

    const float* xs = x + (size_t)series * T_LEN;

#if USE_TDM
    // ---- stage x window with the Tensor Data Mover ----
    // win==0: start at raw idx 112 (keeps TDM pad phase aligned with XI),
    //         threads pre-zero linear [0,119) = all XI(i), i<112 (TDM's first
    //         write is XI(112)=119, so no overlap/race).
    // win>0 : start at raw idx 0, copy full window.
    {
        const int      start_raw = (win == 0) ? HALOP : 0;
        const long long gidx     = (long long)wbase - HALOP + start_raw;   // >= 0
        const unsigned long long gaddr =
            (unsigned long long)(const void*)(xs + gidx);
        const unsigned tile0 = (unsigned)(NXRAW - start_raw);       // elements to move
        const unsigned dim0  = (unsigned)(T_LEN - (int)gidx);       // valid elems; OOB -> 0
        const unsigned lds_base =
            (unsigned)(unsigned long long)(__attribute__((address_space(3))) void*)(void*)lds_x;
        const unsigned lds_off = lds_base + 4u * (unsigned)XI(start_raw);

        // D# group 0: count=1 | lds_addr | global_addr[56:0] | type=2
        u32x4 g0 = { 1u,
                     lds_off,
                     (unsigned)(gaddr & 0xffffffffull),
                     (unsigned)((gaddr >> 32) & 0x01ffffffull) | (2u << 30) };
        // D# group 1: data_size=2 (4B) | pad_enable | pad_interval=3 (16 DW) |
        //             pad_amount=0 (1 DW) | tensor_dim0=dim0 | tile_dim0=tile0
        i32x8 g1 = { (int)((2u << 16) | (1u << 20) | (3u << 22)),
                     (int)((dim0 & 0xffffu) << 16),
                     (int)((dim0 >> 16) & 0xffffu),
                     (int)(tile0 << 16),
                     0, 0, 0, 0 };
        i32x4 gz4 = { 0, 0, 0, 0 };
        i32x8 gz8 = { 0, 0, 0, 0, 0, 0, 0, 0 };

        if (tid < 32) {                              // wave 0 issues the DMA
            __builtin_amdgcn_tensor_load_to_lds(g0, g1, gz4, gz4, gz8, 0);
            __builtin_amdgcn_s_wait_tensorcnt(0);    // TENSORcnt is per-wave
        }
        if (win == 0) {
            for (int i = tid; i < XI(HALOP); i += 256) lds_x[i] = 0.0f;
        }
    }
#else
    // ---- fallback: coalesced global loads into XI-padded LDS ----
    for (int i = tid; i < NXRAW; i += 256) {
        int g = wbase - HALOP + i;
        float v = (g >= 0 && g < T_LEN) ? xs[g] : 0.0f;
        lds_x[XI(i)] = v;
    }
#endif
    __syncthreads();

    // ---- per-wave 256-output Toeplitz WMMA tile ----
    const int wave = tid >> 5;          // 0..7
    const int lane = tid & 31;
    const int half = lane >> 4;         // K sub-select per ISA layout
    const int i16  = lane & 15;         // M for A-matrix, N for B-matrix
    const int tile = wave * 256;        // tile base (relative to wbase)

    v8f acc = {};
    const int kl = 2 * half;            // K_local for VGPR r: kl + r
    #pragma unroll
    for (int kk = 0; kk < 29; ++kk) {
        const int k0 = kk * 4;
        v2f a, b;
        // A[i][k] = wrev[k - i], guard-padded so index 16+k-i in [1,131]
        a.x = lds_w[16 + k0 + kl     - i16];
        a.y = lds_w[16 + k0 + kl + 1 - i16];
        // B[k][n] = |x[t0 - 99 + k + 16 n]| ; raw LDS idx = tile + 13 + k + 16 n
        const int bx = tile + BOFF + k0 + kl + 16 * i16;
        b.x = fabsf(lds_x[XI(bx)]);
        b.y = fabsf(lds_x[XI(bx + 1)]);
        acc = __builtin_amdgcn_wmma_f32_16x16x4_f32(
                  false, a, false, b, (short)0, acc, false, false);
    }

    // ---- epilogue: gain + divide, stage to LDS ----
    #pragma unroll
    for (int r = 0; r < 8; ++r) {
        float Mv = acc[r];
        // D layout: lane half selects M+8, rows r = M, cols = N
        int tl = (r + 8 * half) + 16 * i16 + tile;       // 0..2047
        float xv   = lds_x[XI(HALOP + tl)];
        float gain = powf(fmaxf(Mv, 1e-6f), alpha) + 0.1f;
        lds_o[tl] = xv / gain;
    }
    __syncthreads();

    // ---- coalesced store ----
    float* op = out + (size_t)series * T_LEN + wbase;
    for (int i = tid; i < WIN; i += 256) {
        op[i] = lds_o[i];
    }
}

extern "C" void kernel_launch(void* const* d_in, const int* in_sizes, int n_in,
                              void* d_out, int out_size, void* d_ws, size_t ws_size,
                              hipStream_t stream)
{
    const float* x         = (const float*)d_in[0];
    const float* alpha_raw = (const float*)d_in[1];
    const float* log_s     = (const float*)d_in[2];
    float* out             = (float*)d_out;
    (void)in_sizes; (void)n_in; (void)out_size; (void)d_ws; (void)ws_size;

    dim3 grid(NSERIES * NWIN);   // 16384 blocks: one (series, window) each
    dim3 block(256);             // 8 wave32 waves
    agc_wmma_kernel<<<grid, block, 0, stream>>>(x, alpha_raw, log_s, out);
}